// gcn_layer_87454124081194
// MI455X (gfx1250) — compile-verified
//
#include <hip/hip_runtime.h>

// GCN layer for MI455X (gfx1250, wave32).
// B=8, N=256, D=128, H=8. All GEMM stages use V_WMMA_F32_16X16X4_F32
// (full fp32 matrix path). Each wave register-blocks 4 M-tiles (64x16
// output) so the B fragment is reused across 4 WMMAs per k-step.
// Pairwise edge MLP is factored algebraically:
//   pair @ W_adj  ==  U[b,j] + V[b,i]   (4.3 GFLOP -> 17 MFLOP)
// so the per-edge kernel is pure elementwise / small dense (memory bound).

typedef __attribute__((ext_vector_type(2))) float v2f;
typedef __attribute__((ext_vector_type(8))) float v8f;

#define B_ 8
#define N_ 256
#define D_ 128
#define H_ 8

// One wave computes a 64x16 fp32 block: 4 stacked 16x16 WMMA tiles sharing
// the B fragment each k-step.
// A layout per lane: row = lane&15 (+16*m), float2 at column (k + 2*(lane>>4)).
// B layout per lane: col = lane&15, rows (k + 2*(lane>>4)), (+1).
__device__ __forceinline__ void wmma_block4_f32(const float* __restrict__ A, int lda,
                                                const float* __restrict__ Bm, int ldb,
                                                int K, v8f acc[4]) {
  const int lane = threadIdx.x & 31;
  const int lo   = lane & 15;
  const int hh   = lane >> 4;
  const float* ap = A + lo * lda + 2 * hh;
  const float* bp = Bm + 2 * hh * ldb + lo;
#pragma unroll
  for (int m = 0; m < 4; ++m)
    acc[m] = (v8f){0.f, 0.f, 0.f, 0.f, 0.f, 0.f, 0.f, 0.f};
#pragma unroll 2
  for (int k = 0; k < K; k += 4) {
    v2f b;
    b.x = bp[k * ldb];
    b.y = bp[(k + 1) * ldb];
#pragma unroll
    for (int m = 0; m < 4; ++m) {
      v2f a = *(const v2f*)(ap + m * 16 * lda + k);
      acc[m] = __builtin_amdgcn_wmma_f32_16x16x4_f32(false, a, false, b, (short)0,
                                                     acc[m], false, false);
    }
  }
}

// ---------------- Stage A: support = mol @ weight  [2048x128 x 128x128] ----
__global__ void k_support(const float* __restrict__ mol,
                          const float* __restrict__ weight,
                          float* __restrict__ support) {
  const int wid   = blockIdx.x * (blockDim.x >> 5) + (threadIdx.x >> 5);
  const int mblk  = wid >> 3;   // 0..31  (64 rows each)
  const int ntile = wid & 7;    // 0..7
  const float* A  = mol + (size_t)mblk * 64 * D_;
  const float* Bm = weight + ntile * 16;
  v8f acc[4];
  wmma_block4_f32(A, D_, Bm, D_, D_, acc);
  const int lane = threadIdx.x & 31, lo = lane & 15, hh = lane >> 4;
  const int col = ntile * 16 + lo;
#pragma unroll
  for (int m = 0; m < 4; ++m)
#pragma unroll
    for (int r = 0; r < 8; ++r) {
      const int row = mblk * 64 + m * 16 + r + 8 * hh;
      support[(size_t)row * D_ + col] = acc[m][r];
    }
}

// ---- Stage B: out[b,h] = adj[b,h] @ support[b] + support[b] + bias; tanh;
//               scatter into hbuf[b, i, d*H + h]  (the transpose+reshape). ----
__global__ void k_aggregate(const float* __restrict__ adj,
                            const float* __restrict__ support,
                            const float* __restrict__ bias,
                            const float* __restrict__ mask,
                            float* __restrict__ hbuf) {
  const int wid   = blockIdx.x * (blockDim.x >> 5) + (threadIdx.x >> 5);
  const int bh    = wid >> 5;   // 0..63
  const int rem   = wid & 31;
  const int mblk  = rem >> 3;   // 0..3  (64 rows each)
  const int ntile = rem & 7;    // 0..7
  const int b = bh >> 3, h = bh & 7;
  const float* A  = adj + (size_t)bh * N_ * N_ + (size_t)mblk * 64 * N_;
  const float* Bm = support + (size_t)b * N_ * D_ + ntile * 16;
  v8f acc[4];
  wmma_block4_f32(A, N_, Bm, D_, N_, acc);
  const int lane = threadIdx.x & 31, lo = lane & 15, hh = lane >> 4;
  const int col = ntile * 16 + lo;
  const float bi = bias[col];
#pragma unroll
  for (int m = 0; m < 4; ++m)
#pragma unroll
    for (int r = 0; r < 8; ++r) {
      const int i = mblk * 64 + m * 16 + r + 8 * hh;
      const float sup = support[((size_t)b * N_ + i) * D_ + col];
      const float v = tanhf(acc[m][r] + sup + bi) * mask[b * N_ + i];
      hbuf[((size_t)(b * N_ + i)) * (D_ * H_) + col * H_ + h] = v;
    }
}

// ---- Stage C: out2 = tanh(hbuf @ W_out + b_out) * mask  [2048x1024 x 1024x128]
__global__ void k_out2(const float* __restrict__ hbuf,
                       const float* __restrict__ W_out,
                       const float* __restrict__ b_out,
                       const float* __restrict__ mask,
                       float* __restrict__ out2) {
  const int wid   = blockIdx.x * (blockDim.x >> 5) + (threadIdx.x >> 5);
  const int mblk  = wid >> 3;   // 0..31
  const int ntile = wid & 7;    // 0..7
  const float* A  = hbuf + (size_t)mblk * 64 * (D_ * H_);
  const float* Bm = W_out + ntile * 16;
  v8f acc[4];
  wmma_block4_f32(A, D_ * H_, Bm, D_, D_ * H_, acc);
  const int lane = threadIdx.x & 31, lo = lane & 15, hh = lane >> 4;
  const int col = ntile * 16 + lo;
  const float bo = b_out[col];
#pragma unroll
  for (int m = 0; m < 4; ++m)
#pragma unroll
    for (int r = 0; r < 8; ++r) {
      const int row = mblk * 64 + m * 16 + r + 8 * hh;
      out2[(size_t)row * D_ + col] = tanhf(acc[m][r] + bo) * mask[row];
    }
}

// ---- Stage D: U = out2 @ W_adj[0:128] + b_adj ; V = out2 @ W_adj[128:256] ----
__global__ void k_uv(const float* __restrict__ out2,
                     const float* __restrict__ W_adj,
                     const float* __restrict__ b_adj,
                     float* __restrict__ U,
                     float* __restrict__ V) {
  const int wid  = blockIdx.x * (blockDim.x >> 5) + (threadIdx.x >> 5);
  const int mblk = wid >> 1;   // 0..31
  const int sel  = wid & 1;    // 0 -> U (rows 0:128), 1 -> V (rows 128:256)
  const float* A  = out2 + (size_t)mblk * 64 * D_;
  const float* Bm = W_adj + (size_t)sel * D_ * 16;
  v8f acc[4];
  wmma_block4_f32(A, D_, Bm, 16, D_, acc);
  const int lane = threadIdx.x & 31, lo = lane & 15, hh = lane >> 4;
  float* dst = sel ? V : U;
  const float badd = sel ? 0.f : b_adj[lo];
#pragma unroll
  for (int m = 0; m < 4; ++m)
#pragma unroll
    for (int r = 0; r < 8; ++r) {
      const int row = mblk * 64 + m * 16 + r + 8 * hh;
      dst[(size_t)row * 16 + lo] = acc[m][r] + badd;
    }
}

// ---- Stage E: per edge (b,i,j):
//   t = relu(U[b,j] + V[b,i])                       (b_adj folded into U)
//   a = relu(t @ Wao[0:16] + adj[b,:,i,j] @ Wao[16:24] + b_adjout) * adjmask
//   adj_new[b,h,i,j] = a[h] + adj[b,h,i,j]
// Also passes mask / adjmask through to the output tuple.
__global__ void k_edge(const float* __restrict__ adj,
                       const float* __restrict__ adjmask,
                       const float* __restrict__ U,
                       const float* __restrict__ V,
                       const float* __restrict__ W_adjout,
                       const float* __restrict__ b_adjout,
                       const float* __restrict__ mask,
                       float* __restrict__ out_adj,
                       float* __restrict__ out_mask,
                       float* __restrict__ out_adjmask) {
  __shared__ float Wl[24 * 8];
  __shared__ float bl[8];
  const int tid = threadIdx.x;
  if (tid < 192) Wl[tid] = W_adjout[tid];
  if (tid < 8) bl[tid] = b_adjout[tid];
  __syncthreads();

  const int idx = blockIdx.x * blockDim.x + tid;   // 0 .. B*N*N-1
  const int j = idx & 255;
  const int i = (idx >> 8) & 255;
  const int b = idx >> 16;

  const float* Uj = U + ((size_t)(b * N_ + j)) * 16;
  const float* Vi = V + ((size_t)(b * N_ + i)) * 16;
  float t[16];
#pragma unroll
  for (int c = 0; c < 16; ++c) t[c] = fmaxf(Uj[c] + Vi[c], 0.f);

  float av[8];
#pragma unroll
  for (int h = 0; h < 8; ++h)
    av[h] = adj[(((size_t)(b * H_ + h)) * N_ + i) * N_ + j];

  float a[8];
#pragma unroll
  for (int h = 0; h < 8; ++h) a[h] = bl[h];
#pragma unroll
  for (int c = 0; c < 16; ++c)
#pragma unroll
    for (int h = 0; h < 8; ++h) a[h] = fmaf(t[c], Wl[c * 8 + h], a[h]);
#pragma unroll
  for (int c = 0; c < 8; ++c)
#pragma unroll
    for (int h = 0; h < 8; ++h) a[h] = fmaf(av[c], Wl[(16 + c) * 8 + h], a[h]);

  const float am = adjmask[idx];
#pragma unroll
  for (int h = 0; h < 8; ++h)
    out_adj[(((size_t)(b * H_ + h)) * N_ + i) * N_ + j] =
        fmaxf(a[h], 0.f) * am + av[h];

  out_adjmask[idx] = am;
  if (idx < B_ * N_) out_mask[idx] = mask[idx];
}

extern "C" void kernel_launch(void* const* d_in, const int* in_sizes, int n_in,
                              void* d_out, int out_size, void* d_ws, size_t ws_size,
                              hipStream_t stream) {
  (void)in_sizes; (void)n_in; (void)out_size; (void)ws_size;
  const float* mol      = (const float*)d_in[0];   // [B,N,D]
  const float* adj      = (const float*)d_in[1];   // [B,H,N,N]
  const float* mask     = (const float*)d_in[2];   // [B,N,1]
  const float* adjmask  = (const float*)d_in[3];   // [B,N,N,1]
  const float* weight   = (const float*)d_in[4];   // [D,D]
  const float* bias     = (const float*)d_in[5];   // [D]
  const float* W_out    = (const float*)d_in[6];   // [D*H,D]
  const float* b_out    = (const float*)d_in[7];   // [D]
  const float* W_adj    = (const float*)d_in[8];   // [2D,16]
  const float* b_adj    = (const float*)d_in[9];   // [16]
  const float* W_adjout = (const float*)d_in[10];  // [24,8]
  const float* b_adjout = (const float*)d_in[11];  // [8]

  // Output tuple, concatenated flat: out | adj_new | mask | adjmask
  float* out         = (float*)d_out;                  // B*N*D      = 262144
  float* out_adj     = out + (size_t)B_ * N_ * D_;     // B*H*N*N    = 4194304
  float* out_mask    = out_adj + (size_t)B_ * H_ * N_ * N_;  // B*N  = 2048
  float* out_adjmask = out_mask + (size_t)B_ * N_;     // B*N*N      = 524288

  // Workspace: support | hbuf | U | V   (~9.3 MB total)
  float* ws      = (float*)d_ws;
  float* support = ws;                                  // B*N*D
  float* hbuf    = support + (size_t)B_ * N_ * D_;      // B*N*D*H
  float* U       = hbuf + (size_t)B_ * N_ * D_ * H_;    // B*N*16
  float* V       = U + (size_t)B_ * N_ * 16;            // B*N*16

  // 128 threads = 4 waves per block; one wave per 64x16 output block.
  k_support  <<<64,  128, 0, stream>>>(mol, weight, support);           // 256 waves
  k_aggregate<<<512, 128, 0, stream>>>(adj, support, bias, mask, hbuf); // 2048 waves
  k_out2     <<<64,  128, 0, stream>>>(hbuf, W_out, b_out, mask, out);  // 256 waves
  k_uv       <<<16,  128, 0, stream>>>(out, W_adj, b_adj, U, V);        // 64 waves
  k_edge     <<<2048, 256, 0, stream>>>(adj, adjmask, U, V, W_adjout, b_adjout,
                                        mask, out_adj, out_mask, out_adjmask);
}